// DNATransportGNN_10428180595472
// MI455X (gfx1250) — compile-verified
//
#include <hip/hip_runtime.h>
#include <hip/hip_bf16.h>
#include <math.h>

#define LEAKY 0.2f
#define LN_EPS 1e-5f

typedef float v2f __attribute__((ext_vector_type(2)));
typedef float v8f __attribute__((ext_vector_type(8)));

__device__ __forceinline__ float leaky_relu(float x) { return x >= 0.f ? x : LEAKY * x; }

// sign-aware float atomic max (int max for >=0, uint min for <0)
__device__ __forceinline__ void atomicMaxF(float* addr, float v) {
  if (v >= 0.f) atomicMax((int*)addr, __float_as_int(v));
  else          atomicMin((unsigned int*)addr, __float_as_uint(v));
}

__device__ __forceinline__ float wave_sum(float v) {
  v += __shfl_xor(v, 16, 32);
  v += __shfl_xor(v, 8, 32);
  v += __shfl_xor(v, 4, 32);
  v += __shfl_xor(v, 2, 32);
  v += __shfl_xor(v, 1, 32);
  return v;
}

// ---------------------------------------------------------------- utilities
__global__ void zero_f32(float* p, long n) {
  long t = (long)blockIdx.x * blockDim.x + threadIdx.x;
  if (t < n) p[t] = 0.f;
}

// sum of edge_attr columns (for self-loop mean edge feature)
__global__ void edge_attr_sum(const float* __restrict__ ea, float* __restrict__ sum3, long E) {
  float s0 = 0.f, s1 = 0.f, s2 = 0.f;
  for (long e = (long)blockIdx.x * blockDim.x + threadIdx.x; e < E;
       e += (long)gridDim.x * blockDim.x) {
    s0 += ea[e * 3 + 0]; s1 += ea[e * 3 + 1]; s2 += ea[e * 3 + 2];
  }
  s0 = wave_sum(s0); s1 = wave_sum(s1); s2 = wave_sum(s2);
  if ((threadIdx.x & 31) == 0) {
    atomicAdd(sum3 + 0, s0); atomicAdd(sum3 + 1, s1); atomicAdd(sum3 + 2, s2);
  }
}

// h = x @ node_w + node_b   (K=8)
__global__ void node_embed(const float* __restrict__ x, const float* __restrict__ w,
                           const float* __restrict__ b, float* __restrict__ h, long n) {
  long t = (long)blockIdx.x * blockDim.x + threadIdx.x;
  if (t >= n) return;
  long node = t >> 7; int d = (int)(t & 127);
  float s = b[d];
#pragma unroll
  for (int k = 0; k < 8; k++) s += x[node * 8 + k] * w[k * 128 + d];
  h[t] = s;
}

// ------------------------------------------- WMMA FP32 GEMM:  C[M,128] = A[M,128] @ B[128,128]
// Block = 8 waves: one 128x16 B-panel staged to LDS via async global->LDS copy,
// each wave owns a 16x16 output tile; two independent WMMA accumulation chains.
__global__ void __launch_bounds__(256)
gemm_nk128_wmma(const float* __restrict__ A, const float* __restrict__ B,
                float* __restrict__ C, int M) {
  __shared__ float sB[128 * 16];  // [k][n] row-major, 8 KB
  const int col0 = (blockIdx.x & 7) << 4;   // 8 column tiles of 16
  const int rowS = (blockIdx.x >> 3) << 7;  // 128 rows per block
  const int t = threadIdx.x;

  // ---- stage B panel (128 K x 16 N) into LDS with async copies (ASYNCcnt)
  {
    const float* Bp = B + col0;
#pragma unroll
    for (int it = 0; it < 2; it++) {
      int f = (it * 256 + t) * 4;           // float index 0..2044, 16B chunks
      int k = f >> 4, n = f & 15;
      const float* gp = Bp + (size_t)k * 128 + n;
      unsigned la = (unsigned)(size_t)&sB[k * 16 + n];
      asm volatile("global_load_async_to_lds_b128 %0, %1, off"
                   :: "v"(la), "v"(gp) : "memory");
    }
    asm volatile("s_wait_asynccnt 0x0" ::: "memory");
  }
  __syncthreads();

  const int wv = t >> 5, lane = t & 31;
  const int row0 = rowS + (wv << 4);
  if (row0 >= M) return;

  const int mn = lane & 15;          // m for A, n for B
  const int kh = (lane >> 4) << 1;   // k-half base: 0 or 2
  const float* ap = A + (size_t)(row0 + mn) * 128 + kh;  // A(m, kh + j), b64 contiguous
  const float* bl = sB + kh * 16 + mn;                   // B(kh + j, n) from LDS
  v8f acc0 = {}, acc1 = {};
#pragma unroll
  for (int k = 0; k < 128; k += 8) {
    v2f a0 = { ap[k], ap[k + 1] };
    v2f b0 = { bl[k * 16], bl[k * 16 + 16] };
    acc0 = __builtin_amdgcn_wmma_f32_16x16x4_f32(false, a0, false, b0,
                                                 (short)0, acc0, false, false);
    v2f a1 = { ap[k + 4], ap[k + 5] };
    v2f b1 = { bl[(k + 4) * 16], bl[(k + 4) * 16 + 16] };
    acc1 = __builtin_amdgcn_wmma_f32_16x16x4_f32(false, a1, false, b1,
                                                 (short)0, acc1, false, false);
  }
  v8f acc = acc0 + acc1;
  // D layout: VGPR v -> row = v + 8*(lane>=16), col = lane&15
  int rbase = row0 + ((lane >> 4) << 3);
  int c = col0 + mn;
#pragma unroll
  for (int v = 0; v < 8; v++) C[(size_t)(rbase + v) * 128 + c] = acc[v];
}

// a_src[n,h] = sum_c hw[n,h*32+c]*att_s[h,c];  a_dst likewise
__global__ void node_att_coef(const float* __restrict__ hw, const float* __restrict__ as,
                              const float* __restrict__ ad, float* __restrict__ asrc,
                              float* __restrict__ adst, long n4) {
  long t = (long)blockIdx.x * blockDim.x + threadIdx.x;
  if (t >= n4) return;
  long nd = t >> 2; int h = (int)(t & 3);
  const float* row = hw + nd * 128 + h * 32;
  float s1 = 0.f, s2 = 0.f;
#pragma unroll 8
  for (int c = 0; c < 32; c++) { float v = row[c]; s1 += v * as[h * 32 + c]; s2 += v * ad[h * 32 + c]; }
  asrc[t] = s1; adst[t] = s2;
}

// Per-layer tiny precompute: we[d,h]=sum_c lin_e[d,h*32+c]*a_e[h,c];
// q[k,h]=edge_w[k,:]·we[:,h]; be[h]=edge_b·we[:,h]; selfa[h]=mean3·q[:,h]+be[h]
__global__ void layer_precompute(const float* __restrict__ lin_e, const float* __restrict__ a_e,
                                 const float* __restrict__ edge_w, const float* __restrict__ edge_b,
                                 const float* __restrict__ sum3, float invE,
                                 float* __restrict__ q, float* __restrict__ be,
                                 float* __restrict__ selfa) {
  __shared__ float s_we[128][4];
  __shared__ float s_q[12];
  __shared__ float s_be[4];
  int d = threadIdx.x;  // 128 threads
#pragma unroll
  for (int h = 0; h < 4; h++) {
    float s = 0.f;
    for (int c = 0; c < 32; c++) s += lin_e[d * 128 + h * 32 + c] * a_e[h * 32 + c];
    s_we[d][h] = s;
  }
  __syncthreads();
  if (d < 12) {
    int k = d >> 2, h = d & 3;
    float s = 0.f;
    for (int dd = 0; dd < 128; dd++) s += edge_w[k * 128 + dd] * s_we[dd][h];
    s_q[d] = s; q[d] = s;  // q stored [k][h] = q[k*4+h]
  } else if (d < 16) {
    int h = d & 3;
    float s = 0.f;
    for (int dd = 0; dd < 128; dd++) s += edge_b[dd] * s_we[dd][h];
    s_be[h] = s; be[h] = s;
  }
  __syncthreads();
  if (d < 4) {
    float s = s_be[d];
    for (int k = 0; k < 3; k++) s += sum3[k] * invE * s_q[k * 4 + d];
    selfa[d] = s;
  }
}

// N1: init segment-max with self-loop alpha
__global__ void node_self_max(const float* __restrict__ asrc, const float* __restrict__ adst,
                              const float* __restrict__ selfa, float* __restrict__ m, long n4) {
  long t = (long)blockIdx.x * blockDim.x + threadIdx.x;
  if (t >= n4) return;
  m[t] = leaky_relu(asrc[t] + adst[t] + selfa[t & 3]);
}

// E1: compute edge alpha once (materialized, coalesced) + atomic-max into m
__global__ void edge_alpha_max(const int* __restrict__ src, const int* __restrict__ dst,
                               const float* __restrict__ ea, const float* __restrict__ asrc,
                               const float* __restrict__ adst, const float* __restrict__ q,
                               const float* __restrict__ be, float* __restrict__ alf,
                               float* __restrict__ m, long e4) {
  long t = (long)blockIdx.x * blockDim.x + threadIdx.x;
  if (t >= e4) return;
  int h = (int)(t & 3); long e = t >> 2;
  int s = src[e], d = dst[e];
  float al = asrc[(long)s * 4 + h] + adst[(long)d * 4 + h] +
             ea[e * 3 + 0] * q[h] + ea[e * 3 + 1] * q[4 + h] + ea[e * 3 + 2] * q[8 + h] + be[h];
  al = leaky_relu(al);
  alf[t] = al;
  atomicMaxF(m + (long)d * 4 + h, al);
}

// N2: init denom with self-loop exp
__global__ void node_self_denom(const float* __restrict__ asrc, const float* __restrict__ adst,
                                const float* __restrict__ selfa, const float* __restrict__ m,
                                float* __restrict__ denom, long n4) {
  long t = (long)blockIdx.x * blockDim.x + threadIdx.x;
  if (t >= n4) return;
  float sa = leaky_relu(asrc[t] + adst[t] + selfa[t & 3]);
  denom[t] = expf(sa - m[t]);
}

// E2: atomic-add exp(alpha - m[dst]) into denom (alpha streamed, m gathered)
__global__ void edge_denom(const int* __restrict__ dst, const float* __restrict__ alf,
                           const float* __restrict__ m, float* __restrict__ denom, long e4) {
  long t = (long)blockIdx.x * blockDim.x + threadIdx.x;
  if (t >= e4) return;
  int h = (int)(t & 3); long e = t >> 2;
  int d = dst[e];
  atomicAdd(denom + (long)d * 4 + h, expf(alf[t] - m[(long)d * 4 + h]));
}

// N3: out = self-loop message hw * w_self  (plain store = accumulator init)
__global__ void node_self_msg(const float* __restrict__ hw, const float* __restrict__ asrc,
                              const float* __restrict__ adst, const float* __restrict__ selfa,
                              const float* __restrict__ m, const float* __restrict__ denom,
                              float* __restrict__ outb, long n128) {
  long t = (long)blockIdx.x * blockDim.x + threadIdx.x;
  if (t >= n128) return;
  long n = t >> 7; int dch = (int)(t & 127); int h = dch >> 5;
  float sa = leaky_relu(asrc[n * 4 + h] + adst[n * 4 + h] + selfa[h]);
  float w = expf(sa - m[n * 4 + h]) / (denom[n * 4 + h] + 1e-16f);
  outb[t] = hw[t] * w;
}

// E3: wave-per-edge weighted message scatter; lane covers 4 channels
__global__ void edge_aggregate(const int* __restrict__ src, const int* __restrict__ dst,
                               const float* __restrict__ alf, const float* __restrict__ m,
                               const float* __restrict__ denom, const float* __restrict__ hw,
                               float* __restrict__ outb, long E) {
  long wave = ((long)blockIdx.x * blockDim.x + threadIdx.x) >> 5;
  int lane = threadIdx.x & 31;
  if (wave >= E) return;
  int s = src[wave], d = dst[wave];
  int h = lane >> 3;  // (lane*4)/32
  float al = alf[wave * 4 + h];
  float w = expf(al - m[(long)d * 4 + h]) / (denom[(long)d * 4 + h] + 1e-16f);
  const float4 hv = *(const float4*)(hw + (size_t)s * 128 + lane * 4);
  float* op = outb + (size_t)d * 128 + lane * 4;
  atomicAdd(op + 0, hv.x * w);
  atomicAdd(op + 1, hv.y * w);
  atomicAdd(op + 2, hv.z * w);
  atomicAdd(op + 3, hv.w * w);
}

// N4: h_next = relu(LN(out + bias))  — one wave per node, shuffle reductions
__global__ void node_bias_ln_relu(const float* __restrict__ outb, const float* __restrict__ bias,
                                  const float* __restrict__ gamma, const float* __restrict__ beta,
                                  float* __restrict__ hout, long N) {
  long wave = ((long)blockIdx.x * blockDim.x + threadIdx.x) >> 5;
  int lane = threadIdx.x & 31;
  if (wave >= N) return;
  const float* row = outb + (size_t)wave * 128;
  float v[4]; float s = 0.f;
#pragma unroll
  for (int j = 0; j < 4; j++) { int d = lane * 4 + j; v[j] = row[d] + bias[d]; s += v[j]; }
  s = wave_sum(s);
  float mu = s * (1.f / 128.f);
  float qs = 0.f;
#pragma unroll
  for (int j = 0; j < 4; j++) { float dd = v[j] - mu; qs += dd * dd; }
  qs = wave_sum(qs);
  float rs = rsqrtf(qs * (1.f / 128.f) + LN_EPS);
  float* orow = hout + (size_t)wave * 128;
#pragma unroll
  for (int j = 0; j < 4; j++) {
    int d = lane * 4 + j;
    float y = (v[j] - mu) * rs * gamma[d] + beta[d];
    orow[d] = y > 0.f ? y : 0.f;
  }
}

// pooling: atomic add per (node, channel) + count
__global__ void pool_kernel(const float* __restrict__ h, const int* __restrict__ batch,
                            float* __restrict__ pool, float* __restrict__ cnt, long n128) {
  long t = (long)blockIdx.x * blockDim.x + threadIdx.x;
  if (t >= n128) return;
  long n = t >> 7; int d = (int)(t & 127);
  int g = batch[n];
  atomicAdd(pool + (long)g * 128 + d, h[t]);
  if (d == 0) atomicAdd(cnt + g, 1.f);
}

// fused 2-layer MLP head per graph: out[g,:100] = relu(g@w1+b1)@w2+b2
__global__ void graph_head(const float* __restrict__ pool, const float* __restrict__ cnt,
                           const float* __restrict__ w1, const float* __restrict__ b1,
                           const float* __restrict__ w2, const float* __restrict__ b2,
                           float* __restrict__ outp) {
  __shared__ float gg[128];
  __shared__ float hid[64];
  int g = blockIdx.x, t = threadIdx.x;
  float c = cnt[g]; c = c > 1.f ? c : 1.f;
  gg[t] = pool[(long)g * 128 + t] / c;
  __syncthreads();
  if (t < 64) {
    float s = b1[t];
    for (int d = 0; d < 128; d++) s += gg[d] * w1[d * 64 + t];
    hid[t] = s > 0.f ? s : 0.f;
  }
  __syncthreads();
  if (t < 100) {
    float s = b2[t];
    for (int j = 0; j < 64; j++) s += hid[j] * w2[j * 100 + t];
    outp[(long)g * 100 + t] = s;
  }
}

// ---------------------------------------------------------------- host launch
extern "C" void kernel_launch(void* const* d_in, const int* in_sizes, int n_in,
                              void* d_out, int out_size, void* d_ws, size_t ws_size,
                              hipStream_t stream) {
  (void)n_in; (void)out_size; (void)ws_size;
  const float* x        = (const float*)d_in[0];
  const int*   ei       = (const int*)d_in[1];
  const float* eattr    = (const float*)d_in[2];
  const int*   batch    = (const int*)d_in[3];
  const float* node_w   = (const float*)d_in[4];
  const float* node_b   = (const float*)d_in[5];
  const float* edge_w   = (const float*)d_in[6];
  const float* edge_b   = (const float*)d_in[7];
  const float* gat_lin  = (const float*)d_in[8];
  const float* att_src  = (const float*)d_in[9];
  const float* att_dst  = (const float*)d_in[10];
  const float* lin_edge = (const float*)d_in[11];
  const float* att_edge = (const float*)d_in[12];
  const float* gat_bias = (const float*)d_in[13];
  const float* ln_g     = (const float*)d_in[14];
  const float* ln_b     = (const float*)d_in[15];
  const float* dos_w1   = (const float*)d_in[16];
  const float* dos_b1   = (const float*)d_in[17];
  const float* dos_w2   = (const float*)d_in[18];
  const float* dos_b2   = (const float*)d_in[19];
  const float* trans_w1 = (const float*)d_in[20];
  const float* trans_b1 = (const float*)d_in[21];
  const float* trans_w2 = (const float*)d_in[22];
  const float* trans_b2 = (const float*)d_in[23];

  const long N = in_sizes[0] / 8;
  const long E = in_sizes[2] / 3;
  const int* src = ei;
  const int* dst = ei + E;

  // workspace carve-up (256B aligned)
  char* ws = (char*)d_ws;
  size_t off = 0;
  auto carve = [&](size_t bytes) -> float* {
    float* p = (float*)(ws + off);
    off = (off + bytes + 255) & ~(size_t)255;
    return p;
  };
  float* hbuf  = carve((size_t)N * 128 * 4);
  float* hw    = carve((size_t)N * 128 * 4);
  float* outb  = carve((size_t)N * 128 * 4);
  float* alf   = carve((size_t)E * 4 * 4);
  float* asrc  = carve((size_t)N * 4 * 4);
  float* adst  = carve((size_t)N * 4 * 4);
  float* mbuf  = carve((size_t)N * 4 * 4);
  float* denom = carve((size_t)N * 4 * 4);
  float* sum3  = carve(4 * 4);
  float* qbuf  = carve(12 * 4);
  float* bebuf = carve(4 * 4);
  float* selfa = carve(4 * 4);
  float* pool  = carve(64 * 128 * 4);
  float* cnt   = carve(64 * 4);

  const int BT = 256;
  auto blocks = [](long n, int bt) { return (int)((n + bt - 1) / bt); };

  // once: mean(edge_attr) accumulators + node embedding
  zero_f32<<<1, 32, 0, stream>>>(sum3, 4);
  edge_attr_sum<<<1024, BT, 0, stream>>>(eattr, sum3, E);
  node_embed<<<blocks(N * 128, BT), BT, 0, stream>>>(x, node_w, node_b, hbuf, N * 128);

  // GEMM grid: each block covers 128 rows x 16 cols
  const int row_supers = (int)((N + 127) / 128);
  const int gemm_blocks = row_supers * 8;

  for (int l = 0; l < 4; l++) {
    const float* lin   = gat_lin + (size_t)l * 128 * 128;
    const float* as    = att_src + (size_t)l * 128;
    const float* ad    = att_dst + (size_t)l * 128;
    const float* lin_e = lin_edge + (size_t)l * 128 * 128;
    const float* a_e   = att_edge + (size_t)l * 128;
    const float* bias  = gat_bias + (size_t)l * 128;
    const float* gam   = ln_g + (size_t)l * 128;
    const float* bet   = ln_b + (size_t)l * 128;

    gemm_nk128_wmma<<<gemm_blocks, BT, 0, stream>>>(hbuf, lin, hw, (int)N);
    node_att_coef<<<blocks(N * 4, BT), BT, 0, stream>>>(hw, as, ad, asrc, adst, N * 4);
    layer_precompute<<<1, 128, 0, stream>>>(lin_e, a_e, edge_w, edge_b, sum3,
                                            1.0f / (float)E, qbuf, bebuf, selfa);
    node_self_max<<<blocks(N * 4, BT), BT, 0, stream>>>(asrc, adst, selfa, mbuf, N * 4);
    edge_alpha_max<<<blocks(E * 4, BT), BT, 0, stream>>>(src, dst, eattr, asrc, adst,
                                                         qbuf, bebuf, alf, mbuf, E * 4);
    node_self_denom<<<blocks(N * 4, BT), BT, 0, stream>>>(asrc, adst, selfa, mbuf, denom, N * 4);
    edge_denom<<<blocks(E * 4, BT), BT, 0, stream>>>(dst, alf, mbuf, denom, E * 4);
    node_self_msg<<<blocks(N * 128, BT), BT, 0, stream>>>(hw, asrc, adst, selfa, mbuf,
                                                          denom, outb, N * 128);
    edge_aggregate<<<blocks(E * 32, BT), BT, 0, stream>>>(src, dst, alf, mbuf, denom,
                                                          hw, outb, E);
    node_bias_ln_relu<<<blocks(N * 32, BT), BT, 0, stream>>>(outb, bias, gam, bet, hbuf, N);
  }

  // global mean pool + heads
  zero_f32<<<blocks(64 * 128, BT), BT, 0, stream>>>(pool, 64 * 128);
  zero_f32<<<1, 64, 0, stream>>>(cnt, 64);
  pool_kernel<<<blocks(N * 128, BT), BT, 0, stream>>>(hbuf, batch, pool, cnt, N * 128);

  float* out = (float*)d_out;
  graph_head<<<64, 128, 0, stream>>>(pool, cnt, dos_w1, dos_b1, dos_w2, dos_b2, out);
  graph_head<<<64, 128, 0, stream>>>(pool, cnt, trans_w1, trans_b1, trans_w2, trans_b2,
                                     out + 64 * 100);
}